// GNNBlock_72112500900204
// MI455X (gfx1250) — compile-verified
//
#include <hip/hip_runtime.h>
#include <hip/hip_bf16.h>

#define N_NODES 50000
#define N_EDGES 800000
#define D 64
#define LN_EPS 1e-5f

typedef __attribute__((ext_vector_type(2))) float v2f;
typedef __attribute__((ext_vector_type(8))) float v8f;

// ---------------------------------------------------------------------------
// 1) degree init (self loop contributes 1.0) + zero the LN reduction scalars
// ---------------------------------------------------------------------------
__global__ void deg_init_kernel(float* __restrict__ deg, double* __restrict__ acc, int n) {
    int i = blockIdx.x * blockDim.x + threadIdx.x;
    if (i == 0) { acc[0] = 0.0; acc[1] = 0.0; }
    if (i < n) deg[i] = 1.0f;   // self-loop weight
}

// ---------------------------------------------------------------------------
// 2) deg[col] += w  over edges
// ---------------------------------------------------------------------------
__global__ void deg_accum_kernel(const long long* __restrict__ ei,
                                 const float* __restrict__ ew,
                                 float* __restrict__ deg, int e_cnt) {
    int e = blockIdx.x * blockDim.x + threadIdx.x;
    if (e < e_cnt) {
        int c = (int)ei[e_cnt + e];      // targets = edge_index[1]
        atomicAdd(&deg[c], ew[e]);
    }
}

// ---------------------------------------------------------------------------
// 3) dinv = rsqrt(deg)   (deg >= 1 always because of the self loop)
// ---------------------------------------------------------------------------
__global__ void dinv_kernel(const float* __restrict__ deg, float* __restrict__ dinv, int n) {
    int i = blockIdx.x * blockDim.x + threadIdx.x;
    if (i < n) dinv[i] = rsqrtf(deg[i]);
}

// ---------------------------------------------------------------------------
// 4) h = x @ W  via V_WMMA_F32_16X16X4_F32 (full fp32 precision on matrix pipe)
//    One wave computes a 16x16 tile of h; 16 chained WMMAs cover K=64.
//    Fragment layouts per CDNA5 ISA (wave32):
//      A 16x4 f32 : lane l(0-15) M=l, VGPR0=K0,VGPR1=K1; lanes 16-31 K=2,3
//      B 4x16 f32 : lanes 0-15 N, VGPR0=K0/VGPR1=K1; lanes 16-31 rows K=2,3
//      C/D 16x16  : VGPR i -> M=i (lanes 0-15), M=i+8 (lanes 16-31)
// ---------------------------------------------------------------------------
__global__ void gemm_wmma_kernel(const float* __restrict__ x,
                                 const float* __restrict__ W,
                                 float* __restrict__ h, int n_tiles) {
    int wave = threadIdx.x >> 5;
    int lane = threadIdx.x & 31;
    int tile = blockIdx.x * 8 + wave;
    if (tile >= n_tiles) return;          // wave-uniform: EXEC stays all-ones
    int row0 = (tile >> 2) * 16;          // 4 col-tiles per 16-row strip (D=64)
    int col0 = (tile & 3) * 16;
    int half = lane >> 4;
    int l    = lane & 15;

    v8f c = {0.f, 0.f, 0.f, 0.f, 0.f, 0.f, 0.f, 0.f};
    const float* xrow = x + (row0 + l) * D;
#pragma unroll
    for (int kb = 0; kb < D; kb += 4) {
        int k = kb + 2 * half;
        v2f a, b;
        a.x = xrow[k];
        a.y = xrow[k + 1];
        b.x = W[k * D + col0 + l];
        b.y = W[(k + 1) * D + col0 + l];
        c = __builtin_amdgcn_wmma_f32_16x16x4_f32(
                /*neg_a=*/false, a, /*neg_b=*/false, b,
                /*c_mod=*/(short)0, c, /*reuse_a=*/false, /*reuse_b=*/false);
    }
    float* hp = h + col0 + l;
#pragma unroll
    for (int i = 0; i < 8; ++i)
        hp[(row0 + i + 8 * half) * D] = c[i];
}

// ---------------------------------------------------------------------------
// 5) accumulator init: out[i][d] = h[i][d]*dinv[i]^2 (self loop) + bias[d]
// ---------------------------------------------------------------------------
__global__ void init_out_kernel(const float* __restrict__ h,
                                const float* __restrict__ dinv,
                                const float* __restrict__ bias,
                                float* __restrict__ out, int n) {
    int gid = blockIdx.x * blockDim.x + threadIdx.x;
    int i = gid >> 6;
    int d = gid & 63;
    if (i < n) {
        float di = dinv[i];
        out[gid] = h[gid] * di * di + bias[d];
    }
}

// ---------------------------------------------------------------------------
// 6) edge scatter: out[col][d] += h[row][d] * dinv[row]*w*dinv[col]
//    64 consecutive lanes share one edge -> index/weight loads coalesce into
//    a single L2 request; h gathers and the atomic target are L2-resident.
// ---------------------------------------------------------------------------
__global__ void edge_scatter_kernel(const float* __restrict__ h,
                                    const long long* __restrict__ ei,
                                    const float* __restrict__ ew,
                                    const float* __restrict__ dinv,
                                    float* __restrict__ out, int e_cnt) {
    int gid = blockIdx.x * blockDim.x + threadIdx.x;
    int e = gid >> 6;
    int d = gid & 63;
    if (e >= e_cnt) return;
    int r = (int)ei[e];            // source
    int c = (int)ei[e_cnt + e];    // target
    float norm = dinv[r] * ew[e] * dinv[c];
    atomicAdd(&out[c * D + d], h[r * D + d] * norm);
}

// ---------------------------------------------------------------------------
// 7) global sum / sumsq reduction (graph-mode LayerNorm statistics)
// ---------------------------------------------------------------------------
__global__ void reduce_stats_kernel(const float* __restrict__ x, int n,
                                    double* __restrict__ acc) {
    __shared__ double ssum[256];
    __shared__ double ssq[256];
    double s = 0.0, q = 0.0;
    for (int i = blockIdx.x * blockDim.x + threadIdx.x; i < n;
         i += gridDim.x * blockDim.x) {
        float v = x[i];
        s += (double)v;
        q += (double)v * (double)v;
    }
    ssum[threadIdx.x] = s;
    ssq[threadIdx.x]  = q;
    __syncthreads();
    for (int off = 128; off > 0; off >>= 1) {
        if (threadIdx.x < (unsigned)off) {
            ssum[threadIdx.x] += ssum[threadIdx.x + off];
            ssq[threadIdx.x]  += ssq[threadIdx.x + off];
        }
        __syncthreads();
    }
    if (threadIdx.x == 0) {
        atomicAdd(&acc[0], ssum[0]);
        atomicAdd(&acc[1], ssq[0]);
    }
}

__global__ void stats_final_kernel(const double* __restrict__ acc,
                                   float* __restrict__ mi, double inv_n) {
    double mean = acc[0] * inv_n;
    double var  = acc[1] * inv_n - mean * mean;
    if (var < 0.0) var = 0.0;
    mi[0] = (float)mean;
    mi[1] = (float)(1.0 / (sqrt(var) + (double)LN_EPS));
}

// ---------------------------------------------------------------------------
// 8) (x - mean)/(std+eps) * ln_w + ln_b, then ReLU   (in place on d_out)
// ---------------------------------------------------------------------------
__global__ void finalize_kernel(float* __restrict__ out,
                                const float* __restrict__ mi,
                                const float* __restrict__ lnw,
                                const float* __restrict__ lnb, int total) {
    int gid = blockIdx.x * blockDim.x + threadIdx.x;
    if (gid >= total) return;
    int d = gid & 63;
    float v = (out[gid] - mi[0]) * mi[1] * lnw[d] + lnb[d];
    out[gid] = v > 0.f ? v : 0.f;
}

// ---------------------------------------------------------------------------
extern "C" void kernel_launch(void* const* d_in, const int* in_sizes, int n_in,
                              void* d_out, int out_size, void* d_ws, size_t ws_size,
                              hipStream_t stream) {
    const float*     x    = (const float*)d_in[0];
    const float*     ew   = (const float*)d_in[1];
    const float*     W    = (const float*)d_in[2];
    const float*     bias = (const float*)d_in[3];
    const float*     lnw  = (const float*)d_in[4];
    const float*     lnb  = (const float*)d_in[5];
    const long long* ei   = (const long long*)d_in[6];   // int64 [2, E]

    float* out = (float*)d_out;

    // workspace layout
    char* ws = (char*)d_ws;
    float*  h    = (float*)ws;                                   // N*D floats (12.8 MB)
    float*  deg  = (float*)(ws + (size_t)N_NODES * D * 4);       // N floats
    float*  dinv = deg + N_NODES;                                // N floats
    double* acc  = (double*)(dinv + N_NODES);                    // 2 doubles (8B aligned)
    float*  mi   = (float*)(acc + 2);                            // mean, inv_std

    const int B = 256;
    const int total = N_NODES * D;

    // 1) degree init + zero stats
    deg_init_kernel<<<(N_NODES + B - 1) / B, B, 0, stream>>>(deg, acc, N_NODES);
    // 2) degree accumulation over edges
    deg_accum_kernel<<<(N_EDGES + B - 1) / B, B, 0, stream>>>(ei, ew, deg, N_EDGES);
    // 3) dinv = rsqrt(deg)
    dinv_kernel<<<(N_NODES + B - 1) / B, B, 0, stream>>>(deg, dinv, N_NODES);
    // 4) h = x @ W with fp32 WMMA (N%16==0, D=64 -> 12500 16x16 tiles, 8 waves/block)
    {
        int n_tiles = (N_NODES / 16) * (D / 16);
        int blocks  = (n_tiles + 7) / 8;
        gemm_wmma_kernel<<<blocks, B, 0, stream>>>(x, W, h, n_tiles);
    }
    // 5) out = self-loop term + bias
    init_out_kernel<<<(total + B - 1) / B, B, 0, stream>>>(h, dinv, bias, out, N_NODES);
    // 6) edge scatter-add (E*D threads)
    {
        long long work = (long long)N_EDGES * D;
        int blocks = (int)((work + B - 1) / B);
        edge_scatter_kernel<<<blocks, B, 0, stream>>>(h, ei, ew, dinv, out, N_EDGES);
    }
    // 7) LN statistics over all N*D elements
    reduce_stats_kernel<<<1024, B, 0, stream>>>(out, total, acc);
    stats_final_kernel<<<1, 1, 0, stream>>>(acc, mi, 1.0 / (double)total);
    // 8) normalize + affine + ReLU
    finalize_kernel<<<(total + B - 1) / B, B, 0, stream>>>(out, mi, lnw, lnb, total);
}